// MapLayer_72146860638527
// MI455X (gfx1250) — compile-verified
//
#include <hip/hip_runtime.h>

// out[i] = (idx[i] % 2 == 0) ? sin(x[i]) : cos(x[i])
//
// Memory-bound streaming map: 12 B/element, 402 MB total -> ~17 us at
// 23.3 TB/s. Strategy:
//  - b128 loads and stores (clang ext vectors) for full cacheline efficiency
//  - non-temporal hints: working set (402 MB) > L2 (192 MB), pure streaming
//  - single hardware TRANS op per element via cos(x) = sin(x + pi/2):
//      v_sin_f32 takes pre-scaled input (revolutions), so
//      t = fma(x, 1/2pi, odd ? 0.25f : 0.0f); r = v_sin_f32(t)
//    Inputs are N(0,1) so |t| < ~1: well inside the accurate range.

typedef float v4f __attribute__((ext_vector_type(4)));
typedef int   v4i __attribute__((ext_vector_type(4)));

#define INV_2PI 0.15915494309189535f

__device__ __forceinline__ float sincos_sel(float x, int idx) {
    // odd index -> add quarter revolution (pi/2 in radians) -> cos
    float bias = (idx & 1) ? 0.25f : 0.0f;
    float t = __builtin_fmaf(x, INV_2PI, bias);
    return __builtin_amdgcn_sinf(t);   // raw v_sin_f32 (input in revolutions)
}

__global__ __launch_bounds__(256) void map_sincos_vec4(
    const v4f* __restrict__ x4,
    const v4i* __restrict__ i4,
    v4f*       __restrict__ o4,
    int n4)
{
    int i = blockIdx.x * blockDim.x + threadIdx.x;
    if (i >= n4) return;

    v4f v  = __builtin_nontemporal_load(&x4[i]);
    v4i id = __builtin_nontemporal_load(&i4[i]);

    v4f r;
    r.x = sincos_sel(v.x, id.x);
    r.y = sincos_sel(v.y, id.y);
    r.z = sincos_sel(v.z, id.z);
    r.w = sincos_sel(v.w, id.w);

    __builtin_nontemporal_store(r, &o4[i]);
}

// Scalar tail (not taken for 4096*8192, kept for generality)
__global__ __launch_bounds__(64) void map_sincos_tail(
    const float* __restrict__ x,
    const int*   __restrict__ idx,
    float*       __restrict__ out,
    int start, int n)
{
    int i = start + blockIdx.x * blockDim.x + threadIdx.x;
    if (i >= n) return;
    out[i] = sincos_sel(x[i], idx[i]);
}

extern "C" void kernel_launch(void* const* d_in, const int* in_sizes, int n_in,
                              void* d_out, int out_size, void* d_ws, size_t ws_size,
                              hipStream_t stream)
{
    const float* x   = (const float*)d_in[0];
    const int*   idx = (const int*)d_in[1];
    float*       out = (float*)d_out;

    int n  = out_size;        // 4096 * 8192 = 33,554,432
    int n4 = n >> 2;          // 128-bit vectors
    int rem_start = n4 << 2;

    if (n4 > 0) {
        const int block = 256;
        int grid = (n4 + block - 1) / block;
        map_sincos_vec4<<<grid, block, 0, stream>>>(
            (const v4f*)x, (const v4i*)idx, (v4f*)out, n4);
    }
    if (rem_start < n) {
        int tail = n - rem_start;
        map_sincos_tail<<<(tail + 63) / 64, 64, 0, stream>>>(
            x, idx, out, rem_start, n);
    }
}